// MoELoRALinearLayer_50878182588815
// MI455X (gfx1250) — compile-verified
//
#include <hip/hip_runtime.h>

// MoE-LoRA fused kernel for gfx1250 (MI455X), wave32.
// down = H @ Wd^T  -> topk scale/scatter mask -> out = R @ Wu^T
// N=16384, D_IN=D_OUT=4096, RANK=64, TOPK=8. All fp32 -> V_WMMA_F32_16X16X4_F32.
// Phase-1 staging uses double-buffered GLOBAL_LOAD_ASYNC_TO_LDS_B128 (ASYNCcnt).

typedef __attribute__((ext_vector_type(2))) float v2f;
typedef __attribute__((ext_vector_type(8))) float v8f;

#define NTOK 16384
#define DIN  4096
#define DOUT 4096
#define RANK 64
#define TOPK 8
#define MT   64      // token rows per block
#define KC   64      // K-chunk for the down GEMM
#define PAD  68      // LDS row stride (floats): 16B-aligned rows + conflict-free b64 lane reads

static __device__ __forceinline__ v8f wmma_f32(v2f a, v2f b, v8f c) {
    // 8 args: (neg_a, A, neg_b, B, c_mod, C, reuse_a, reuse_b)
    return __builtin_amdgcn_wmma_f32_16x16x4_f32(false, a, false, b, (short)0, c, false, false);
}

// Async global->LDS copy, 16 bytes per lane. vdst = per-lane LDS byte address,
// vaddr = per-lane 64-bit global address (GV mode). Tracked by ASYNCcnt.
static __device__ __forceinline__ void async_copy_b128(unsigned lds_byte_addr,
                                                       const float* gaddr) {
    asm volatile("global_load_async_to_lds_b128 %0, %1, off"
                 :: "v"(lds_byte_addr), "v"(gaddr)
                 : "memory");
}

// Low 32 bits of a flat __shared__ pointer are the wave-relative LDS address.
static __device__ __forceinline__ unsigned lds_addr32(const void* p) {
    return (unsigned)(uintptr_t)p;
}

__global__ __launch_bounds__(256)
void moe_lora_fused(const float* __restrict__ hidden,
                    const float* __restrict__ down_w,
                    const float* __restrict__ up_w,
                    const float* __restrict__ tkv,
                    const long long* __restrict__ tki,
                    float* __restrict__ out)
{
    __shared__ float Ash[2][MT][PAD];    // hidden chunks (double buffered); Ash[0] reused as mask
    __shared__ float Bsh[2][RANK][PAD];  // down_w chunks (double buffered)
    __shared__ float Rsh[MT][PAD];       // masked down output = phase-2 A matrix

    const int t    = threadIdx.x;
    const int wave = t >> 5;
    const int lane = t & 31;
    const int half = lane >> 4;   // 0: K pair {0,1} / M rows 0..7 ; 1: K pair {2,3} / M rows 8..15
    const int l    = lane & 15;
    const int row0 = blockIdx.x * MT;

    // Per-thread staging coordinates: 4 float4 tiles of the 64x64 chunk, coalesced.
    // it-th tile: flat = it*1024 + t*4 -> row = flat>>6, col = flat&63.

    // Phase-1 tile assignment: 64x64 down output = 4x4 grid of 16x16 tiles.
    // wave w owns tiles 2w and 2w+1 (same M sub-tile -> shared A operand).
    const int mi  = wave >> 1;
    const int m0  = mi * 16;
    const int n0a = ((2 * wave) & 3) * 16;
    const int n0b = n0a + 16;

    v8f c0 = {}; v8f c1 = {};

    // Kick off async staging of chunk 0 into buffer 0 (8 async ops per thread).
    #pragma unroll
    for (int it = 0; it < 4; ++it) {
        const int f = it * 1024 + t * 4;
        const int r = f >> 6, cc = f & 63;
        async_copy_b128(lds_addr32(&Ash[0][r][cc]), hidden + (size_t)(row0 + r) * DIN + cc);
        async_copy_b128(lds_addr32(&Bsh[0][r][cc]), down_w + (size_t)r * DIN + cc);
    }

    // ---------------- Phase 1: down = H[64 x 4096] @ Wd^T ----------------
    for (int kb = 0; kb < DIN; kb += KC) {
        const int cur = (kb / KC) & 1;
        if (kb + KC < DIN) {
            const int nxt = cur ^ 1;
            #pragma unroll
            for (int it = 0; it < 4; ++it) {
                const int f = it * 1024 + t * 4;
                const int r = f >> 6, cc = f & 63;
                async_copy_b128(lds_addr32(&Ash[nxt][r][cc]),
                                hidden + (size_t)(row0 + r) * DIN + kb + KC + cc);
                async_copy_b128(lds_addr32(&Bsh[nxt][r][cc]),
                                down_w + (size_t)r * DIN + kb + KC + cc);
            }
            // Retire current chunk's 8 async ops (in-order completion); next 8 in flight.
            asm volatile("s_wait_asynccnt 8" ::: "memory");
        } else {
            asm volatile("s_wait_asynccnt 0" ::: "memory");
        }
        __syncthreads();   // all waves' async writes to buffer `cur` visible

        #pragma unroll
        for (int kk = 0; kk < KC; kk += 4) {
            // A(m,k): lanes 0-15 -> K {kk,kk+1}; lanes 16-31 -> K {kk+2,kk+3}
            v2f a  = *(const v2f*)&Ash[cur][m0  + l][kk + 2 * half];
            v2f b0 = *(const v2f*)&Bsh[cur][n0a + l][kk + 2 * half];
            v2f b1 = *(const v2f*)&Bsh[cur][n0b + l][kk + 2 * half];
            c0 = wmma_f32(a, b0, c0);
            c1 = wmma_f32(a, b1, c1);
        }
        __syncthreads();   // buffer `cur` free for the prefetch issued at iteration kb + KC
    }

    // ---------------- Top-k gather/scale/scatter as a dense mask ----------------
    float (*Msk)[PAD] = Ash[0];
    for (int i = t; i < MT * RANK; i += 256) Msk[i >> 6][i & 63] = 0.0f;
    __syncthreads();
    if (t < MT) {
        const long long* ip = tki + (size_t)(row0 + t) * TOPK;
        const float*     vp = tkv + (size_t)(row0 + t) * TOPK;
        #pragma unroll
        for (int j = 0; j < TOPK; ++j) Msk[t][(int)ip[j]] = vp[j];
    }
    __syncthreads();

    // Apply mask to accumulators (D layout: c[v] = (M=m0+v+8*half, N=n0+l)) -> Rsh
    #pragma unroll
    for (int v = 0; v < 8; ++v) {
        const int mr = m0 + v + 8 * half;
        Rsh[mr][n0a + l] = c0[v] * Msk[mr][n0a + l];
        Rsh[mr][n0b + l] = c1[v] * Msk[mr][n0b + l];
    }
    __syncthreads();

    // ---------------- Phase 2: out[64 x 4096] = Rsh[64 x 64] @ Wu^T ----------------
    // wave -> M sub-tile (2 waves each), parity splits the 256 N sub-tiles.
    const int m0p = (wave >> 1) * 16;

    // Hoist the entire K=64 A operand into registers once (16 x v2f = 32 VGPRs).
    v2f Areg[16];
    #pragma unroll
    for (int k4 = 0; k4 < 16; ++k4)
        Areg[k4] = *(const v2f*)&Rsh[m0p + l][4 * k4 + 2 * half];

    for (int nt = 0; nt < 128; ++nt) {
        const int n0p = (((nt << 1) | (wave & 1)) << 4);
        v8f acc = {};
        // B(k=r, n=o) = up_w[o][r]; lane supplies column o = n0p + l (L2-resident, 1 MB total)
        const float* bp = up_w + (size_t)(n0p + l) * RANK + 2 * half;
        #pragma unroll
        for (int k4 = 0; k4 < 16; ++k4) {
            v2f b = *(const v2f*)(bp + 4 * k4);
            acc = wmma_f32(Areg[k4], b, acc);
        }
        // out is 256 MB written exactly once: stream it (NT hint), keep L2 for weights.
        float* op = out + (size_t)(row0 + m0p + 8 * half) * DOUT + n0p + l;
        #pragma unroll
        for (int v = 0; v < 8; ++v)
            __builtin_nontemporal_store(acc[v], op + (size_t)v * DOUT);
    }
}

extern "C" void kernel_launch(void* const* d_in, const int* in_sizes, int n_in,
                              void* d_out, int out_size, void* d_ws, size_t ws_size,
                              hipStream_t stream) {
    (void)in_sizes; (void)n_in; (void)out_size; (void)d_ws; (void)ws_size;
    const float*     hidden = (const float*)d_in[0];
    const float*     down_w = (const float*)d_in[1];
    const float*     up_w   = (const float*)d_in[2];
    const float*     tkv    = (const float*)d_in[3];
    const long long* tki    = (const long long*)d_in[4];  // int64 indices
    float* out = (float*)d_out;

    dim3 grid(NTOK / MT);   // 256 blocks
    dim3 block(256);        // 8 waves (wave32)
    hipLaunchKernelGGL(moe_lora_fused, grid, block, 0, stream,
                       hidden, down_w, up_w, tkv, tki, out);
}